// Net_76854144794846
// MI455X (gfx1250) — compile-verified
//
#include <hip/hip_runtime.h>
#include <math.h>

typedef __attribute__((ext_vector_type(2))) float v2f;
typedef __attribute__((ext_vector_type(8))) float v8f;

#define GALPHA 0.6f
#define GINV   (1.0f / (1.0f + 0.6f))
#define KSTEPS 4
#define SCAN_B 256

// ---------------------------------------------------------------------------
// Dual GEMM with fp32 WMMA:  outc = A*Wc + bc,  outl = A*Wl + bl
// A: [N, Cin] row-major, W: [Cin, Cout] row-major.
// Block = (Cout/16) waves covering a 32-row strip (2 row tiles per wave, so
// each B load feeds two v_wmma). A tile (32 x Cin) staged in LDS via the
// CDNA5 async global->LDS path (ASYNCcnt). TCIN/TCOUT = 0 -> runtime dims.
// ---------------------------------------------------------------------------
template<int TCIN, int TCOUT>
__global__ void gemm_dual_wmma(const float* __restrict__ A,
                               const float* __restrict__ Wc, const float* __restrict__ bc,
                               const float* __restrict__ Wl, const float* __restrict__ bl,
                               float* __restrict__ outc, float* __restrict__ outl,
                               int N, int nCin, int nCout)
{
    const int Cin  = TCIN  ? TCIN  : nCin;
    const int Cout = TCOUT ? TCOUT : nCout;
    extern __shared__ float lds[];                 // 32 * Cin floats
    const int wave = threadIdx.x >> 5;
    const int lane = threadIdx.x & 31;
    const int half = lane >> 4;                    // 0: lanes 0-15, 1: lanes 16-31
    const int l16  = lane & 15;
    const int row0 = blockIdx.x * 32;

    // Async global->LDS staging of the A tile: 16 bytes per lane per issue.
    // Out-of-range tile rows are EXEC-masked off; their (uninitialized) LDS
    // rows only feed D rows that are never stored (A row m -> D row m only).
    {
        const int elems = 32 * Cin;                // floats in tile
        const unsigned lbase = (unsigned)(size_t)&lds[0];
        const unsigned long long gbase =
            (unsigned long long)(const void*)A + (unsigned long long)row0 * Cin * 4ull;
        for (int i4 = threadIdx.x * 4; i4 < elems; i4 += blockDim.x * 4) {
            int r = i4 / Cin;                      // tile row (i4 stays within one row)
            if (row0 + r < N) {
                unsigned loff = lbase + (unsigned)i4 * 4u;
                unsigned goff = (unsigned)i4 * 4u;
                asm volatile("global_load_async_to_lds_b128 %0, %1, %2"
                             :: "v"(loff), "v"(goff), "s"(gbase)
                             : "memory");
            }
        }
        asm volatile("s_wait_asynccnt 0x0" ::: "memory");
    }
    __syncthreads();

    const int col0 = wave * 16;
    const float* a0 = lds + l16 * Cin;             // rows 0..15 of strip
    const float* a1 = lds + (16 + l16) * Cin;      // rows 16..31 of strip

    v8f cc0 = {}, cc1 = {}, ll0 = {}, ll1 = {};

    #pragma unroll 4
    for (int k0 = 0; k0 < Cin; k0 += 4) {
        const int k = k0 + 2 * half;               // ISA A layout: VGPR v = K = 2*half + v
        v2f av0; av0.x = a0[k]; av0.y = a0[k + 1];
        v2f av1; av1.x = a1[k]; av1.y = a1[k + 1];
        v2f bcv, blv;                              // ISA B layout mirrored: B[2*half+v][l16]
        bcv.x = Wc[(size_t)k       * Cout + col0 + l16];
        bcv.y = Wc[(size_t)(k + 1) * Cout + col0 + l16];
        blv.x = Wl[(size_t)k       * Cout + col0 + l16];
        blv.y = Wl[(size_t)(k + 1) * Cout + col0 + l16];
        cc0 = __builtin_amdgcn_wmma_f32_16x16x4_f32(false, av0, false, bcv, (short)0, cc0, false, false);
        cc1 = __builtin_amdgcn_wmma_f32_16x16x4_f32(false, av1, false, bcv, (short)0, cc1, false, false);
        ll0 = __builtin_amdgcn_wmma_f32_16x16x4_f32(false, av0, false, blv, (short)0, ll0, false, false);
        ll1 = __builtin_amdgcn_wmma_f32_16x16x4_f32(false, av1, false, blv, (short)0, ll1, false, false);
    }

    const float biasc = bc[col0 + l16];
    const float biasl = bl[col0 + l16];
    #pragma unroll
    for (int r = 0; r < 8; ++r) {                  // C layout: VGPR r -> row M = r + 8*half
        int rowa = row0 + r + 8 * half;
        int rowb = rowa + 16;
        if (rowa < N) {
            size_t o = (size_t)rowa * Cout + col0 + l16;
            outc[o] = cc0[r] + biasc;
            outl[o] = ll0[r] + biasl;
        }
        if (rowb < N) {
            size_t o = (size_t)rowb * Cout + col0 + l16;
            outc[o] = cc1[r] + biasc;
            outl[o] = ll1[r] + biasl;
        }
    }
}

// ---------------------------------------------------------------------------
// CSR build: histogram -> exclusive scan -> scatter fill (+edge norms)
// ---------------------------------------------------------------------------
__global__ void zero_i(int* __restrict__ p, long long n)
{
    long long i = (long long)blockIdx.x * blockDim.x + threadIdx.x;
    if (i < n) p[i] = 0;
}

__global__ void cnt_edges(const long long* __restrict__ dst, int* __restrict__ cnt, long long E)
{
    long long e = (long long)blockIdx.x * blockDim.x + threadIdx.x;
    if (e < E) atomicAdd(&cnt[(int)dst[e]], 1);    // global_atomic_add_u32
}

__global__ void dinv_from_cnt(const int* __restrict__ cnt, float* __restrict__ dinv, long long n)
{
    long long i = (long long)blockIdx.x * blockDim.x + threadIdx.x;
    if (i < n) dinv[i] = rsqrtf((float)(cnt[i] + 1));   // +1 self loop; deg >= 1
}

__global__ void scan_blocks(const int* __restrict__ in, int* __restrict__ out,
                            int* __restrict__ sums, int n)
{
    __shared__ int s[SCAN_B];
    int i = blockIdx.x * SCAN_B + threadIdx.x;
    int v = (i < n) ? in[i] : 0;
    s[threadIdx.x] = v;
    __syncthreads();
    for (int off = 1; off < SCAN_B; off <<= 1) {
        int t = (threadIdx.x >= off) ? s[threadIdx.x - off] : 0;
        __syncthreads();
        s[threadIdx.x] += t;
        __syncthreads();
    }
    if (i < n) out[i] = s[threadIdx.x] - v;        // exclusive
    if (threadIdx.x == SCAN_B - 1) sums[blockIdx.x] = s[threadIdx.x];
}

__global__ void scan_sums(int* __restrict__ sums, int nb)   // single block, nb <= 1024
{
    __shared__ int s[1024];
    int v = (threadIdx.x < nb) ? sums[threadIdx.x] : 0;
    s[threadIdx.x] = v;
    __syncthreads();
    for (int off = 1; off < 1024; off <<= 1) {
        int t = (threadIdx.x >= off) ? s[threadIdx.x - off] : 0;
        __syncthreads();
        s[threadIdx.x] += t;
        __syncthreads();
    }
    if (threadIdx.x < nb) sums[threadIdx.x] = s[threadIdx.x] - v;   // exclusive
}

__global__ void scan_add(int* __restrict__ out, const int* __restrict__ sums, int n)
{
    int i = blockIdx.x * SCAN_B + threadIdx.x;
    if (i < n) out[i] += sums[blockIdx.x];
}

__global__ void fill_csr(const long long* __restrict__ src, const long long* __restrict__ dst,
                         const int* __restrict__ rowptr, int* __restrict__ cursor,
                         int* __restrict__ col, float* __restrict__ wedge,
                         const float* __restrict__ dinv, long long E)
{
    long long e = (long long)blockIdx.x * blockDim.x + threadIdx.x;
    if (e >= E) return;
    int s = (int)src[e];
    int d = (int)dst[e];
    int pos = rowptr[d] + atomicAdd(&cursor[d], 1);
    col[pos]   = s;
    wedge[pos] = dinv[s] * dinv[d];                // edge norm precomputed once
}

// ---------------------------------------------------------------------------
// Fused diffusion step (gather SpMM + update, no atomics):
//   hout = (xc + alpha * (sum_j w_j * hin[col_j] + dinv^2 * hin)) * inv
// One thread = 4 channels of one node; for C=128 a wave owns one node, so the
// edge loop is wave-uniform. shift = log2(C/4).
// ---------------------------------------------------------------------------
__global__ void spmm_gather_update(const int* __restrict__ rowptr, const int* __restrict__ cnt,
                                   const int* __restrict__ col, const float* __restrict__ wedge,
                                   const float* __restrict__ dinv,
                                   const float* __restrict__ xc,
                                   const float* __restrict__ hin,
                                   float* __restrict__ hout,
                                   int N, int C, int shift)
{
    long long t = (long long)blockIdx.x * blockDim.x + threadIdx.x;
    int node = (int)(t >> shift);
    if (node >= N) return;
    int c0 = (int)(t & ((1 << shift) - 1)) << 2;

    const int start = rowptr[node];
    const int num   = cnt[node];

    float4 acc = make_float4(0.f, 0.f, 0.f, 0.f);
    for (int j = 0; j < num; ++j) {
        int   s = col[start + j];
        float w = wedge[start + j];
        const float4 hv = *(const float4*)(hin + (size_t)s * C + c0);
        acc.x += w * hv.x; acc.y += w * hv.y; acc.z += w * hv.z; acc.w += w * hv.w;
    }
    const float di = dinv[node];
    const float sw = di * di;                      // self-loop weight
    const float4 xv = *(const float4*)(xc  + (size_t)node * C + c0);
    const float4 hv = *(const float4*)(hin + (size_t)node * C + c0);
    float4 o;
    o.x = (xv.x + GALPHA * (acc.x + sw * hv.x)) * GINV;
    o.y = (xv.y + GALPHA * (acc.y + sw * hv.y)) * GINV;
    o.z = (xv.z + GALPHA * (acc.z + sw * hv.z)) * GINV;
    o.w = (xv.w + GALPHA * (acc.w + sw * hv.w)) * GINV;
    *(float4*)(hout + (size_t)node * C + c0) = o;
}

// out = elu(h + xl) or (h + xl);  safe when out aliases xl.
__global__ void combine_kernel(const float* __restrict__ h, const float* __restrict__ xl,
                               float* __restrict__ out, long long total, int do_elu)
{
    long long i = (long long)blockIdx.x * blockDim.x + threadIdx.x;
    if (i >= total) return;
    float v = h[i] + xl[i];
    if (do_elu) v = (v > 0.0f) ? v : (__expf(v) - 1.0f);
    out[i] = v;
}

// ---------------------------------------------------------------------------
// Host side
// ---------------------------------------------------------------------------
static inline unsigned blocks_for(long long n, int t) { return (unsigned)((n + t - 1) / t); }

static void launch_gemm(const float* A, const float* Wc, const float* bc,
                        const float* Wl, const float* bl,
                        float* outc, float* outl, int N, int Cin, int Cout,
                        hipStream_t stream)
{
    dim3 blk(32u * (unsigned)(Cout / 16));
    unsigned grd = blocks_for(N, 32);
    size_t ldsz = (size_t)32 * Cin * sizeof(float);
    if (Cin == 256 && Cout == 128)
        gemm_dual_wmma<256,128><<<grd, blk, ldsz, stream>>>(A, Wc, bc, Wl, bl, outc, outl, N, Cin, Cout);
    else if (Cin == 128 && Cout == 128)
        gemm_dual_wmma<128,128><<<grd, blk, ldsz, stream>>>(A, Wc, bc, Wl, bl, outc, outl, N, Cin, Cout);
    else if (Cin == 128 && Cout == 64)
        gemm_dual_wmma<128,64><<<grd, blk, ldsz, stream>>>(A, Wc, bc, Wl, bl, outc, outl, N, Cin, Cout);
    else
        gemm_dual_wmma<0,0><<<grd, blk, ldsz, stream>>>(A, Wc, bc, Wl, bl, outc, outl, N, Cin, Cout);
}

static void run_layer(const float* in, int Cin, int Cout,
                      const float* Wc, const float* bc, const float* Wl, const float* bl,
                      const int* rowptr, const int* cnt, const int* col, const float* wedge,
                      const float* dinv,
                      float* xc, float* xl, float* hA, float* hB,
                      float* out, int do_elu, long long N, hipStream_t stream)
{
    launch_gemm(in, Wc, bc, Wl, bl, xc, xl, (int)N, Cin, Cout, stream);

    int logC = 0; while ((1 << logC) < Cout) ++logC;
    const int shift = logC - 2;                    // threads per node = Cout/4
    const long long total   = N * (long long)Cout;
    const long long gthread = N << shift;

    const float* hin = xc;                         // h_0 = xc
    float* hout = hA;
    for (int k = 0; k < KSTEPS; ++k) {
        spmm_gather_update<<<blocks_for(gthread, 256), 256, 0, stream>>>(
            rowptr, cnt, col, wedge, dinv, xc, hin, hout, (int)N, Cout, shift);
        hin  = hout;
        hout = (hout == hA) ? hB : hA;
    }
    combine_kernel<<<blocks_for(total, 256), 256, 0, stream>>>(hin, xl, out, total, do_elu);
}

extern "C" void kernel_launch(void* const* d_in, const int* in_sizes, int n_in,
                              void* d_out, int out_size, void* d_ws, size_t ws_size,
                              hipStream_t stream)
{
    const float*     x  = (const float*)d_in[0];
    const long long* ei = (const long long*)d_in[1];
    const float* Wc1 = (const float*)d_in[2];  const float* bc1 = (const float*)d_in[3];
    const float* Wl1 = (const float*)d_in[4];  const float* bl1 = (const float*)d_in[5];
    const float* Wc2 = (const float*)d_in[6];  const float* bc2 = (const float*)d_in[7];
    const float* Wl2 = (const float*)d_in[8];  const float* bl2 = (const float*)d_in[9];
    const float* Wc3 = (const float*)d_in[10]; const float* bc3 = (const float*)d_in[11];
    const float* Wl3 = (const float*)d_in[12]; const float* bl3 = (const float*)d_in[13];

    const int H    = in_sizes[3];                  // 128
    const int Cin  = in_sizes[2] / H;              // 256
    const int Cout = in_sizes[11];                 // 64
    const long long N = (long long)in_sizes[0] / Cin;   // 50000
    const long long E = (long long)in_sizes[1] / 2;     // 800000

    const long long* src = ei;
    const long long* dst = ei + E;

    // ---- workspace carve -------------------------------------------------
    auto al = [](size_t b) { return (b + 255) & ~(size_t)255; };
    const size_t NBf = al((size_t)N * H * sizeof(float));
    const size_t Ni  = al((size_t)N * sizeof(int));
    const size_t Ei  = al((size_t)E * sizeof(int));
    const int    nScanBlocks = (int)((N + SCAN_B - 1) / SCAN_B);

    char* p = (char*)d_ws;
    float* bA   = (float*)p; p += NBf;             // xc
    float* bB   = (float*)p; p += NBf;             // xl (layers 1,3) / layer1 out
    float* bC   = (float*)p; p += NBf;             // h ping
    float* bD   = (float*)p; p += NBf;             // h pong
    float* bE   = (float*)p; p += NBf;             // xl layer2 / layer2 out
    float* dinv = (float*)p; p += Ni;
    int* cnt    = (int*)p;   p += Ni;
    int* rowptr = (int*)p;   p += Ni;
    int* cursor = (int*)p;   p += Ni;
    int* col    = (int*)p;   p += Ei;
    float* wedge= (float*)p; p += Ei;
    int* sums   = (int*)p;   p += al((size_t)nScanBlocks * sizeof(int));
    (void)ws_size; (void)n_in; (void)out_size;

    // ---- CSR build + normalization (fresh every call) --------------------
    zero_i<<<blocks_for(N, 256), 256, 0, stream>>>(cnt, N);
    cnt_edges<<<blocks_for(E, 256), 256, 0, stream>>>(dst, cnt, E);
    dinv_from_cnt<<<blocks_for(N, 256), 256, 0, stream>>>(cnt, dinv, N);
    scan_blocks<<<nScanBlocks, SCAN_B, 0, stream>>>(cnt, rowptr, sums, (int)N);
    scan_sums<<<1, 1024, 0, stream>>>(sums, nScanBlocks);
    scan_add<<<nScanBlocks, SCAN_B, 0, stream>>>(rowptr, sums, (int)N);
    zero_i<<<blocks_for(N, 256), 256, 0, stream>>>(cursor, N);
    fill_csr<<<blocks_for(E, 256), 256, 0, stream>>>(src, dst, rowptr, cursor, col, wedge, dinv, E);

    // ---- layers ----------------------------------------------------------
    // L1: x[N,256] -> [N,128], ELU, out into bB (in-place combine over xl)
    run_layer(x,  Cin, H, Wc1, bc1, Wl1, bl1, rowptr, cnt, col, wedge, dinv,
              bA, bB, bC, bD, bB, /*elu=*/1, N, stream);
    // L2: bB -> [N,128], ELU, out into bE
    run_layer(bB, H,   H, Wc2, bc2, Wl2, bl2, rowptr, cnt, col, wedge, dinv,
              bA, bE, bC, bD, bE, /*elu=*/1, N, stream);
    // L3: bE -> [N,64], no ELU, straight to d_out
    run_layer(bE, H, Cout, Wc3, bc3, Wl3, bl3, rowptr, cnt, col, wedge, dinv,
              bA, bB, bC, bD, (float*)d_out, /*elu=*/0, N, stream);
}